// YOLOLoss_6416681140369
// MI455X (gfx1250) — compile-verified
//
#include <hip/hip_runtime.h>

typedef __attribute__((ext_vector_type(2))) float v2f;
typedef __attribute__((ext_vector_type(8))) float v8f;

#define SGRID 7
#define NCLS 20
#define CELLS_TOTAL (16384 * 49)   // 802816
#define BLOCK 256
#define NBLOCKS (CELLS_TOTAL / BLOCK)  // 3136 exactly

// ---------------------------------------------------------------------------
// Wave32 reduction via V_WMMA_F32_16X16X4_F32.
// A (16x4 f32, 2 VGPRs/lane): lane L<16 -> A[L][0]=v, A[L][1]=0
//                             lane L>=16 -> A[L-16][2]=v, A[L-16][3]=0
// B (4x16) = all ones  =>  D[m][n] = A[m][0]+A[m][1]+A[m][2]+A[m][3]
// Each lane 0..15 holds D[0..7][lane] in 8 VGPRs, lanes 16..31 hold D[8..15][*].
// Summing the 8 D registers gives (per lane group) the half-wave total; one
// shfl_down(16) completes the full wave sum (valid in lanes 0..15).
// Requires EXEC == all ones (guaranteed: exact-fit launch, reconverged point).
// ---------------------------------------------------------------------------
__device__ __forceinline__ float wave_reduce_wmma(float v) {
    v2f a;
    a.x = v;
    a.y = 0.0f;
    v2f b;
    b.x = 1.0f;
    b.y = 1.0f;
    v8f c = {};
    v8f d = __builtin_amdgcn_wmma_f32_16x16x4_f32(
        /*neg_a=*/false, a, /*neg_b=*/false, b,
        /*c_mod=*/(short)0, c, /*reuse_a=*/false, /*reuse_b=*/false);
    float s = 0.0f;
#pragma unroll
    for (int i = 0; i < 8; ++i) s += d[i];
    // lanes 0..15 hold sum of rows 0..7, lanes 16..31 hold sum of rows 8..15
    s += __shfl_down(s, 16, 32);
    return s;  // full wave total in lanes 0..15 (we use lane 0)
}

// Block (256 threads = 8 waves) reduction; result valid on threadIdx.x == 0.
__device__ __forceinline__ float block_reduce(float v) {
    __shared__ float wsum[8];
    float w = wave_reduce_wmma(v);
    const int lane = threadIdx.x & 31;
    const int wid  = threadIdx.x >> 5;
    if (lane == 0) wsum[wid] = w;
    __syncthreads();
    float t = 0.0f;
    if (threadIdx.x == 0) {
#pragma unroll
        for (int i = 0; i < 8; ++i) t += wsum[i];
    }
    return t;
}

// ---------------------------------------------------------------------------
// Main kernel: one thread per (batch, i, j) cell. Streaming NT b64 loads.
// ---------------------------------------------------------------------------
__global__ __launch_bounds__(BLOCK) void yolo_loss_main(
    const float* __restrict__ preds, const float* __restrict__ labels,
    float* __restrict__ partial) {
    const int cell = blockIdx.x * BLOCK + threadIdx.x;

    // cell = ((batch*7 + i)*7 + j):  grid_x = j, grid_y = i
    const int j = cell % SGRID;
    const int i = (cell / SGRID) % SGRID;
    const float gx = (float)j;
    const float gy = (float)i;

    const v2f* p2 = reinterpret_cast<const v2f*>(preds) + (size_t)cell * 15;
    const v2f* l2 = reinterpret_cast<const v2f*>(labels) + (size_t)cell * 15;

    // Pairs 0..4: boxes + confidences (elements 0..9)
    float pb[10], lb[10];
#pragma unroll
    for (int k = 0; k < 5; ++k) {
        v2f tp = __builtin_nontemporal_load(p2 + k);
        v2f tl = __builtin_nontemporal_load(l2 + k);
        pb[2 * k] = tp.x; pb[2 * k + 1] = tp.y;
        lb[2 * k] = tl.x; lb[2 * k + 1] = tl.y;
    }
    // Pairs 5..14: class probabilities (elements 10..29), consumed on the fly
    float cls_err = 0.0f;
#pragma unroll
    for (int k = 5; k < 15; ++k) {
        v2f tp = __builtin_nontemporal_load(p2 + k);
        v2f tl = __builtin_nontemporal_load(l2 + k);
        float d0 = tp.x - tl.x;
        float d1 = tp.y - tl.y;
        cls_err += d0 * d0 + d1 * d1;
    }

    const float inv_s = 1.0f / 7.0f;
    const float lx = lb[0], ly = lb[1], lw = lb[2], lh = lb[3];
    const float obj = lb[4];

    const float lcx = (gx + lx) * inv_s;
    const float lcy = (gy + ly) * inv_s;
    const float lminx = lcx - 0.5f * lw, lmaxx = lcx + 0.5f * lw;
    const float lminy = lcy - 0.5f * lh, lmaxy = lcy + 0.5f * lh;
    const float area_l = lw * lh;
    const float slw = sqrtf(lw), slh = sqrtf(lh);

    float iou[2], conf[2], box_err[2];
#pragma unroll
    for (int b = 0; b < 2; ++b) {
        const int o = 5 * b;
        const float px = pb[o], py = pb[o + 1], pw = pb[o + 2], ph = pb[o + 3];
        conf[b] = pb[o + 4];
        const float pcx = (gx + px) * inv_s;
        const float pcy = (gy + py) * inv_s;
        const float pminx = pcx - 0.5f * pw, pmaxx = pcx + 0.5f * pw;
        const float pminy = pcy - 0.5f * ph, pmaxy = pcy + 0.5f * ph;
        float iw = fminf(pmaxx, lmaxx) - fmaxf(pminx, lminx);
        float ih = fminf(pmaxy, lmaxy) - fmaxf(pminy, lminy);
        iw = fmaxf(iw, 0.0f);
        ih = fmaxf(ih, 0.0f);
        const float inter = iw * ih;
        iou[b] = inter / (pw * ph + area_l - inter + 1e-10f);
        const float dx = px - lx;
        const float dy = py - ly;
        const float dw = sqrtf(pw) - slw;
        const float dh = sqrtf(ph) - slh;
        box_err[b] = dx * dx + dy * dy + dw * dw + dh * dh;
    }

    // argmax with tie -> index 0 (jnp.argmax semantics)
    const float r0 = (iou[0] >= iou[1]) ? 1.0f : 0.0f;
    const float resp[2] = {r0, 1.0f - r0};

    float loss = 0.0f;
#pragma unroll
    for (int b = 0; b < 2; ++b) {
        const float orsp = obj * resp[b];
        loss += 5.0f * orsp * box_err[b];              // LAMBDA_COORD
        const float dc = conf[b] - iou[b];
        loss += orsp * dc * dc;                        // conf (obj)
        loss += 0.5f * (1.0f - orsp) * conf[b] * conf[b];  // LAMBDA_NOOBJ
    }
    loss += obj * cls_err;

    const float tot = block_reduce(loss);
    if (threadIdx.x == 0) partial[blockIdx.x] = tot;
}

// ---------------------------------------------------------------------------
// Final deterministic fold of the 3136 block partials.
// ---------------------------------------------------------------------------
__global__ __launch_bounds__(BLOCK) void yolo_loss_final(
    const float* __restrict__ partial, float* __restrict__ out) {
    float s = 0.0f;
    for (int i = threadIdx.x; i < NBLOCKS; i += BLOCK) s += partial[i];
    const float tot = block_reduce(s);
    if (threadIdx.x == 0) out[0] = tot * (1.0f / 16384.0f);
}

extern "C" void kernel_launch(void* const* d_in, const int* in_sizes, int n_in,
                              void* d_out, int out_size, void* d_ws, size_t ws_size,
                              hipStream_t stream) {
    const float* preds  = (const float*)d_in[0];
    const float* labels = (const float*)d_in[1];
    float* partial = (float*)d_ws;  // 3136 floats
    yolo_loss_main<<<NBLOCKS, BLOCK, 0, stream>>>(preds, labels, partial);
    yolo_loss_final<<<1, BLOCK, 0, stream>>>(partial, (float*)d_out);
}